// LocalCosineSimilarity_22660247453704
// MI455X (gfx1250) — compile-verified
//
#include <hip/hip_runtime.h>

typedef __attribute__((ext_vector_type(2))) float v2f;
typedef __attribute__((ext_vector_type(8))) float v8f;

#define BATCH 8
#define CH 3
#define H_ 1024
#define W_ 1024
#define KW 11
#define RAD 5
#define EPS_ 1e-6f

#define TILE_H 64
#define TILE_W 32
#define HALO_H (TILE_H + KW - 1)   /* 74 */
#define HALO_W (TILE_W + KW - 1)   /* 42 */
#define SPAD (HALO_W + 2)          /* 44: LDS row stride */
#define HPAD (TILE_W + 2)          /* 34 */

// One separable-box-filter pass for a single product plane:
//   horizontal K=11 sliding sum (VALU) -> LDS, then
//   vertical K=11 sum as Band(16x26) x HS(26x16) via chained fp32 WMMA.
__device__ __forceinline__ v8f box_pass(const float (&ss)[HALO_H][SPAD],
                                        float (&sh)[HALO_H][HPAD],
                                        const v2f (&band)[7],
                                        int tid, int rb, int cb, int half, int l15)
{
    __syncthreads();   // prior writers of ss / prior readers of sh are done

    // Horizontal sliding box sum: one thread per halo row.
    for (int lr = tid; lr < HALO_H; lr += 256) {
        float run = 0.f;
#pragma unroll
        for (int i = 0; i < KW; ++i) run += ss[lr][i];
        sh[lr][0] = run;
        for (int c = 1; c < TILE_W; ++c) {
            run += ss[lr][c + KW - 1] - ss[lr][c - 1];
            sh[lr][c] = run;
        }
    }
    __syncthreads();

    // Vertical box sum on the matrix core, 7 K-chunks of 4.
    v8f c8 = {0.f, 0.f, 0.f, 0.f, 0.f, 0.f, 0.f, 0.f};
#pragma unroll
    for (int t = 0; t < 7; ++t) {
        int k0 = 4 * t + 2 * half;             // this lane's K pair in the chunk
        int kb0 = k0     < 26 ? k0     : 25;   // clamp: band is 0 there anyway
        int kb1 = k0 + 1 < 26 ? k0 + 1 : 25;
        v2f bv;
        bv.x = sh[rb + kb0][cb + l15];         // B: lane holds data col l15
        bv.y = sh[rb + kb1][cb + l15];
        c8 = __builtin_amdgcn_wmma_f32_16x16x4_f32(
                 false, band[t], false, bv, (short)0, c8, false, false);
    }
    return c8;
}

__global__ __launch_bounds__(256)
void LocalCosineSimilarity_22660247453704_kernel(const float* __restrict__ x,
                                                 const float* __restrict__ y,
                                                 float* __restrict__ out)
{
    __shared__ float s_s[3][HALO_H][SPAD];   // product planes xy, xx, yy (39,072 B)
    __shared__ float s_h[HALO_H][HPAD];      // reused hsum plane       (10,064 B)

    const int tid     = threadIdx.x;
    const int tile_c0 = blockIdx.x * TILE_W;
    const int tile_r0 = blockIdx.y * TILE_H;
    const int b       = blockIdx.z;

    const long plane = (long)H_ * W_;
    const float* xb = x + (long)b * CH * plane;
    const float* yb = y + (long)b * CH * plane;

    // ---- Phase 1: coalesced load + channel reduce into LDS product planes ----
    for (int p = tid; p < HALO_H * HALO_W; p += 256) {
        int lr = p / HALO_W;
        int lc = p - lr * HALO_W;
        int gr = tile_r0 - RAD + lr;
        int gc = tile_c0 - RAD + lc;
        float xx = 0.f, yy = 0.f, xy = 0.f;
        if (gr >= 0 && gr < H_ && gc >= 0 && gc < W_) {
            long idx = (long)gr * W_ + gc;
#pragma unroll
            for (int c = 0; c < CH; ++c) {
                float xv = xb[c * plane + idx];
                float yv = yb[c * plane + idx];
                xx = fmaf(xv, xv, xx);
                yy = fmaf(yv, yv, yy);
                xy = fmaf(xv, yv, xy);
            }
        }
        s_s[0][lr][lc] = xy;   // zero-padded OOB == 'SAME' zero padding
        s_s[1][lr][lc] = xx;
        s_s[2][lr][lc] = yy;
    }

    const int wave = tid >> 5;
    const int lane = tid & 31;
    const int half = lane >> 4;   // 0: lanes 0-15, 1: lanes 16-31
    const int l15  = lane & 15;
    const int rb   = (wave >> 1) * 16;   // sub-tile origin (halo-row coords)
    const int cb   = (wave & 1) * 16;

    // Band matrix A (Band[m][k] = 1 iff m <= k <= m+10), computed once,
    // statically indexed -> stays in 14 VGPRs across all three passes.
    v2f band[7];
#pragma unroll
    for (int t = 0; t < 7; ++t) {
        int k0 = 4 * t + 2 * half;
        band[t].x = (k0     >= l15 && k0     <= l15 + 10) ? 1.f : 0.f;
        band[t].y = (k0 + 1 >= l15 && k0 + 1 <= l15 + 10) ? 1.f : 0.f;
    }

    // Three fully-register-resident accumulators (no arrays -> no scratch).
    v8f accXY = box_pass(s_s[0], s_h, band, tid, rb, cb, half, l15);
    v8f accXX = box_pass(s_s[1], s_h, band, tid, rb, cb, half, l15);
    v8f accYY = box_pass(s_s[2], s_h, band, tid, rb, cb, half, l15);

    // ---- Phase 4: cosine similarity + store ----
    // D layout: VGPR e -> row e (lanes 0-15) / row e+8 (lanes 16-31), col = l15.
    const int orow0 = tile_r0 + rb;
    const int ocol  = tile_c0 + cb + l15;
    float* ob = out + (long)b * plane;
#pragma unroll
    for (int e = 0; e < 8; ++e) {
        int mrow = e + half * 8;
        float v = accXY[e] / (__builtin_sqrtf(accXX[e]) * __builtin_sqrtf(accYY[e]) + EPS_);
        ob[(long)(orow0 + mrow) * W_ + ocol] = v;
    }
}

extern "C" void kernel_launch(void* const* d_in, const int* in_sizes, int n_in,
                              void* d_out, int out_size, void* d_ws, size_t ws_size,
                              hipStream_t stream) {
    const float* x = (const float*)d_in[0];
    const float* y = (const float*)d_in[1];
    float* out = (float*)d_out;
    dim3 grid(W_ / TILE_W, H_ / TILE_H, BATCH);   // (32, 16, 8)
    LocalCosineSimilarity_22660247453704_kernel<<<grid, dim3(256), 0, stream>>>(x, y, out);
}